// ROIAlignPooler_69329362092397
// MI455X (gfx1250) — compile-verified
//
#include <hip/hip_runtime.h>

// ---------------------------------------------------------------------------
// 3D FPN ROI-Align for MI455X (gfx1250, wave32).
// Separable formulation: out = Bz @ (By @ (Bx @ F)) per ROI, with the
// dominant x-contraction done with V_WMMA_F32_16X16X4_F32 (fp32 WMMA, same
// arithmetic class as the fp32 reference).
// Features are staged channel-last in d_ws (~37.4 MB) so the 64-channel axis
// is contiguous; the whole pyramid lives in the 192MB L2, so pass-1 B-operand
// loads come from L2 at full rate.
// ---------------------------------------------------------------------------

typedef __attribute__((ext_vector_type(2))) float v2f;
typedef __attribute__((ext_vector_type(8))) float v8f;

#define CC     64     // channels
#define NROI   256    // 2 batches * 128 boxes
#define SMAX   20     // K padded to multiple of 4 for WMMA
#define SREG   17     // max real span of gathered indices per dim
#define CTILE  16     // channel tile (WMMA N)

// ---- Kernel 0: NCDHW -> NDHWC transpose of one pyramid level ---------------
__global__ void roi3d_transpose_kernel(const float* __restrict__ src,
                                       float* __restrict__ dst, int dhw) {
  int tid = blockIdx.x * blockDim.x + threadIdx.x;
  int n = 2 * CC * dhw;
  if (tid >= n) return;
  int c = tid % CC;
  int s = (tid / CC) % dhw;
  int b = tid / (CC * dhw);
  dst[tid] = src[((long)b * CC + c) * dhw + s];   // coalesced writes
}

// ---- Kernel 1: per-(roi, ctile) separable ROI-Align with WMMA pass-1 -------
__global__ __launch_bounds__(256)
void roi3d_pool_kernel(const float* __restrict__ boxes,
                       const float* __restrict__ f0,
                       const float* __restrict__ f1,
                       const float* __restrict__ f2,
                       const float* __restrict__ f3,
                       float* __restrict__ out) {
  __shared__ float Bx [16 * SMAX];        // WMMA A matrix, rows 7..15 zero
  __shared__ float Byz[2 * 7 * SMAX];     // [0]=Bz rows, [1]=By rows
  __shared__ float T1s[SREG * 7 * CTILE]; // x-contracted slice  [ry][px][c]
  __shared__ float T2 [SREG * 7 * 7 * CTILE]; // y-contracted   [rz][py][px][c]
  __shared__ int   sI[8];                 // level, Dn, start[3], span[3]
  __shared__ float sScale;

  const int roi  = blockIdx.x;
  const int c0   = blockIdx.y * CTILE;
  const int tid  = threadIdx.x;
  const float* box = boxes + (long)roi * 6;   // (z1,y1,x1,z2,y2,x2)

  // Phase 0: level selection (reference: clip(floor(4+log2(sqrt(vol)/224+eps)),2,5)-2)
  if (tid == 0) {
    float area = (box[3] - box[0] + 1.0f) * (box[4] - box[1] + 1.0f) *
                 (box[5] - box[2] + 1.0f);
    float s  = sqrtf(area);
    float lf = floorf(4.0f + log2f(s / 224.0f + 1e-6f));
    lf = fminf(fmaxf(lf, 2.0f), 5.0f);
    int level = (int)lf - 2;
    sI[0] = level;
    sI[1] = 40 >> level;                 // cubic feature dims 40/20/10/5
    sScale = 0.25f / (float)(1 << level);
  }
  __syncthreads();

  const int   Dn    = sI[1];
  const float scale = sScale;

  // Phase A: per-dim region start/span; zero the interpolation matrices
  for (int i = tid; i < 16 * SMAX + 2 * 7 * SMAX; i += blockDim.x) {
    if (i < 16 * SMAX) Bx[i] = 0.0f; else Byz[i - 16 * SMAX] = 0.0f;
  }
  if (tid < 3) {
    int d = tid;
    float lo  = box[d] * scale;
    float len = fmaxf(box[3 + d] * scale - lo, 1.0f);
    float bin = len * (1.0f / 7.0f);
    float cA = lo + 0.25f * bin;                       // first sample (s=0)
    float cB = lo + 6.75f * bin;                       // last sample  (s=13)
    int li0  = (int)floorf(fminf(fmaxf(cA, 0.0f), (float)(Dn - 1)));
    int li13 = (int)floorf(fminf(fmaxf(cB, 0.0f), (float)(Dn - 1)));
    int hi13 = min(li13 + 1, Dn - 1);
    sI[2 + d] = li0;
    sI[5 + d] = min(hi13 - li0 + 1, SREG);             // span <= 17
  }
  __syncthreads();

  // Phase B: build dense interp matrices (weights fold the 1/8 sample mean,
  // the `valid` mask, and border clamping; zero-padded K columns annihilate
  // any defensively-clamped loads in pass 1)
  if (tid < 21) {
    int d  = tid / 7;                                  // 0=z 1=y 2=x
    int px = tid % 7;
    float lo  = box[d] * scale;
    float len = fmaxf(box[3 + d] * scale - lo, 1.0f);
    float bin = len * (1.0f / 7.0f);
    int start = sI[2 + d], span = sI[5 + d];
    float* row = (d == 2) ? &Bx[px * SMAX] : &Byz[d * 7 * SMAX + px * SMAX];
    for (int t = 0; t < 2; ++t) {
      int   s   = 2 * px + t;
      float pos = ((float)s + 0.5f) * 0.5f;            // RATIO=2
      float c   = lo + pos * bin;
      float valid = (c > -1.0f && c < (float)Dn) ? 1.0f : 0.0f;
      float cc  = fminf(fmaxf(c, 0.0f), (float)(Dn - 1));
      float low = floorf(cc);
      int   li  = (int)low;
      int   hi  = min(li + 1, Dn - 1);
      float whi = (cc - low) * valid * 0.5f;           // 0.5 = per-dim mean
      float wlo = (1.0f - (cc - low)) * valid * 0.5f;
      int cl = li - start, ch = hi - start;
      if (cl >= 0 && cl < span) row[cl] += wlo;
      if (ch >= 0 && ch < span) row[ch] += whi;
    }
  }
  __syncthreads();

  const int level = sI[0];
  const int zs = sI[2], ys = sI[3], xs = sI[4];
  const int spz = sI[5], spy = sI[6], spx = sI[7]; (void)spx;
  const float* F = (level == 0) ? f0 : (level == 1) ? f1 : (level == 2) ? f2 : f3;
  const long  bbase = (long)(roi >> 7) * (long)Dn * Dn * Dn * CC;

  const int wave = tid >> 5;
  const int lane = tid & 31;
  const int half = lane >> 4;
  const int lm   = lane & 15;
  const float* Bz = &Byz[0];
  const float* By = &Byz[7 * SMAX];

  for (int rz = 0; rz < spz; ++rz) {
    const int zi = zs + rz;

    // ---- Pass 1 (WMMA): T1[ry][px][c] = sum_x Bx[px][x] * F[zi][yi][x][c]
    // A (16x4 f32): VGPR0 = K(k0+0 | k0+2), VGPR1 = K(k0+1 | k0+3), M = lane&15
    // B (4x16 f32): VGPR0 = rows k0+0|k0+1, VGPR1 = rows k0+2|k0+3, N = lane&15
    for (int ry = wave; ry < spy; ry += 8) {           // wave-uniform trip count
      const int yi = ys + ry;
      const long rowb = bbase + ((long)(zi * Dn + yi) * Dn) * CC + c0 + lm;
      v8f acc = {0.f, 0.f, 0.f, 0.f, 0.f, 0.f, 0.f, 0.f};
      for (int kk = 0; kk < 5; ++kk) {
        const int k0 = kk * 4;
        v2f a, b;
        a[0] = Bx[lm * SMAX + k0 + (half ? 2 : 0)];
        a[1] = Bx[lm * SMAX + k0 + (half ? 3 : 1)];
        const int x0i = min(xs + k0 + half,     Dn - 1); // pad cols: weight==0
        const int x1i = min(xs + k0 + 2 + half, Dn - 1);
        b[0] = F[rowb + (long)x0i * CC];
        b[1] = F[rowb + (long)x1i * CC];
        acc = __builtin_amdgcn_wmma_f32_16x16x4_f32(
            false, a, false, b, (short)0, acc, false, false);
      }
      // D layout: VGPR i -> M=i (lanes 0-15) / M=i+8 (lanes 16-31); keep M<7
      if (half == 0) {
        for (int i = 0; i < 7; ++i)
          T1s[(ry * 7 + i) * CTILE + lm] = acc[i];
      }
    }
    __syncthreads();

    // ---- Pass 2 (VALU): T2[rz][py][px][c] = sum_ry By[py][ry] * T1[ry][px][c]
    for (int i = tid; i < 7 * 7 * CTILE; i += blockDim.x) {
      const int c = i & (CTILE - 1);
      const int p = i >> 4;
      const int px = p % 7, py = p / 7;
      float a2 = 0.0f;
      for (int ry = 0; ry < spy; ++ry)
        a2 = fmaf(By[py * SMAX + ry], T1s[(ry * 7 + px) * CTILE + c], a2);
      T2[(rz * 49 + py * 7 + px) * CTILE + c] = a2;
    }
    __syncthreads();                                   // T1s reused next slice
  }

  // ---- Pass 3 (VALU): out[c][pz][py][px] = sum_rz Bz[pz][rz] * T2[...]
  for (int i = tid; i < 343 * CTILE; i += blockDim.x) {
    const int c = i & (CTILE - 1);
    const int p = i >> 4;
    const int px = p % 7, py = (p / 7) % 7, pz = p / 49;
    float a3 = 0.0f;
    for (int rz = 0; rz < spz; ++rz)
      a3 = fmaf(Bz[pz * SMAX + rz], T2[(rz * 49 + py * 7 + px) * CTILE + c], a3);
    out[((long)roi * CC + c0 + c) * 343 + (pz * 49 + py * 7 + px)] = a3;
  }
}

// ---------------------------------------------------------------------------
extern "C" void kernel_launch(void* const* d_in, const int* in_sizes, int n_in,
                              void* d_out, int out_size, void* d_ws, size_t ws_size,
                              hipStream_t stream) {
  const float* x0 = (const float*)d_in[0];   // (2,64,40,40,40)
  const float* x1 = (const float*)d_in[1];   // (2,64,20,20,20)
  const float* x2 = (const float*)d_in[2];   // (2,64,10,10,10)
  const float* x3 = (const float*)d_in[3];   // (2,64,5,5,5)
  const float* bx = (const float*)d_in[4];   // (2,128,6)
  float* out = (float*)d_out;                // (256,64,7,7,7)

  // channel-last staging areas in workspace (float offsets); total 37.44 MB
  float* w  = (float*)d_ws;
  float* t0 = w;                 // 2*64*64000
  float* t1 = w + 8192000;       // 2*64*8000
  float* t2 = w + 9216000;       // 2*64*1000
  float* t3 = w + 9344000;       // 2*64*125   (end 9360000 floats)

  const int dhw[4] = {64000, 8000, 1000, 125};
  const float* src[4] = {x0, x1, x2, x3};
  float* dst[4] = {t0, t1, t2, t3};
  for (int l = 0; l < 4; ++l) {
    int n = 2 * CC * dhw[l];
    roi3d_transpose_kernel<<<(n + 255) / 256, 256, 0, stream>>>(src[l], dst[l], dhw[l]);
  }

  dim3 grid(NROI, CC / CTILE, 1);            // (256 rois, 4 channel tiles)
  roi3d_pool_kernel<<<grid, 256, 0, stream>>>(bx, t0, t1, t2, t3, out);
}